// ComplexMultiheadAttention_31903017074958
// MI455X (gfx1250) — compile-verified
//
#include <hip/hip_runtime.h>

// Problem constants (from the reference)
#define B_ 2
#define L_ 2048
#define E_ 1024
#define H_ 16
#define D_ 64   // E/H

typedef float v2f __attribute__((ext_vector_type(2)));
typedef float v8f __attribute__((ext_vector_type(8)));

// D = A(16x4, f32) * B(4x16, f32) + C(16x16, f32)
// 8-arg form: (neg_a, A, neg_b, B, c_mod, C, reuse_a, reuse_b); f32 requires neg_a=neg_b=0.
__device__ __forceinline__ v8f wmma_f32(v2f a, v2f b, v8f c) {
    return __builtin_amdgcn_wmma_f32_16x16x4_f32(false, a, false, b, (short)0, c,
                                                 false, false);
}

// ---------------------------------------------------------------------------
// Complex linear: yr = xr@Wr^T - xi@Wi^T + br ; yi = xr@Wi^T + xi@Wr^T + bi
// One wave (32 threads) computes a 16x64 output tile (both yr and yi):
// A fragments (xr/xi/-xi) are reused across 4 B column-tiles -> 16 WMMAs per
// 10 float2 loads per k-step.
// x: [M, E] row-major, W: [E, E] row-major (used transposed), M = B*L.
// grid = (E/64, M/16), block = 32.
// ---------------------------------------------------------------------------
__global__ __launch_bounds__(32)
void complex_linear_wmma(const float* __restrict__ xr, const float* __restrict__ xi,
                         const float* __restrict__ Wr, const float* __restrict__ Wi,
                         const float* __restrict__ br, const float* __restrict__ bi,
                         float* __restrict__ yr, float* __restrict__ yi)
{
    const int lane = threadIdx.x & 31;
    const int nl   = lane & 15;   // A: row m ; B: col n ; C: col n
    const int hf   = lane >> 4;   // half-wave selects K pair (A/B) / row half (C)
    const int mt   = blockIdx.y;
    const int nt   = blockIdx.x;  // 64-wide column supertile

    // A fragments: x[m][k0 + 2*hf + {0,1}]  (contiguous float2 per lane)
    const float* xr_p = xr + (size_t)(mt * 16 + nl) * E_ + 2 * hf;
    const float* xi_p = xi + (size_t)(mt * 16 + nl) * E_ + 2 * hf;
    // B fragments: B(k,n) = W[n][k] -> W[n][k0 + 2*hf + {0,1}]; the 4 column
    // tiles are j*16*E floats apart (folds into the 24-bit inst offset).
    const float* wr_p = Wr + (size_t)(nt * 64 + nl) * E_ + 2 * hf;
    const float* wi_p = Wi + (size_t)(nt * 64 + nl) * E_ + 2 * hf;

    v8f cr[4], ci[4];
#pragma unroll
    for (int j = 0; j < 4; ++j) {
        cr[j] = (v8f){0.f, 0.f, 0.f, 0.f, 0.f, 0.f, 0.f, 0.f};
        ci[j] = (v8f){0.f, 0.f, 0.f, 0.f, 0.f, 0.f, 0.f, 0.f};
    }

#pragma unroll 2
    for (int k0 = 0; k0 < E_; k0 += 4) {
        v2f ar  = *(const v2f*)(xr_p + k0);
        v2f ai  = *(const v2f*)(xi_p + k0);
        v2f nai = -ai;                 // f32 WMMA has no A-neg modifier
        v2f bwr[4], bwi[4];
#pragma unroll
        for (int j = 0; j < 4; ++j) {
            bwr[j] = *(const v2f*)(wr_p + (size_t)j * 16 * E_ + k0);
            bwi[j] = *(const v2f*)(wi_p + (size_t)j * 16 * E_ + k0);
        }
#pragma unroll
        for (int j = 0; j < 4; ++j) {
            cr[j] = wmma_f32(ar,  bwr[j], cr[j]);   // + xr * Wr^T
            cr[j] = wmma_f32(nai, bwi[j], cr[j]);   // - xi * Wi^T
            ci[j] = wmma_f32(ar,  bwi[j], ci[j]);   // + xr * Wi^T
            ci[j] = wmma_f32(ai,  bwr[j], ci[j]);   // + xi * Wr^T
        }
    }

#pragma unroll
    for (int j = 0; j < 4; ++j) {
        const int   n   = nt * 64 + j * 16 + nl;
        const float bbr = br[n];
        const float bbi = bi[n];
#pragma unroll
        for (int g = 0; g < 8; ++g) {
            const int row = mt * 16 + g + 8 * hf;
            yr[(size_t)row * E_ + n] = cr[j][g] + bbr;
            yi[(size_t)row * E_ + n] = ci[j][g] + bbi;
        }
    }
}

// ---------------------------------------------------------------------------
// Flash attention over complex q/k/v (real softmax of Re(q conj(k)^T)).
// One wave handles 16 query rows of one (b, h); streams key tiles of 16,
// keeping scores + online-softmax state + 16x64 r/i output accums in VGPRs.
// Heads are read/written in place inside the [B, L, E] buffers (cols h*D..).
// grid = (L/16, B*H), block = 32.
// ---------------------------------------------------------------------------
__global__ __launch_bounds__(32)
void cattn_flash_wmma(const float* __restrict__ Qr, const float* __restrict__ Qi,
                      const float* __restrict__ Kr, const float* __restrict__ Ki,
                      const float* __restrict__ Vr, const float* __restrict__ Vi,
                      float* __restrict__ Or, float* __restrict__ Oi)
{
    __shared__ float lds[16 * 16];   // P tile bounce buffer (C-layout -> A-layout)

    const int lane = threadIdx.x & 31;
    const int nl   = lane & 15;
    const int hf   = lane >> 4;
    const int mt   = blockIdx.x;               // query tile
    const int bh   = blockIdx.y;
    const int b    = bh / H_;
    const int h    = bh % H_;
    const size_t rowbase = (size_t)b * L_ * E_;
    const int    hcol    = h * D_;
    const float  scale   = 0.125f;             // 1/sqrt(D)

    // Preload Q tile fragments: 16 K-steps over D=64, A-layout float2 per lane.
    v2f qa_r[16], qa_i[16];
    {
        const float* qr_p = Qr + rowbase + (size_t)(mt * 16 + nl) * E_ + hcol + 2 * hf;
        const float* qi_p = Qi + rowbase + (size_t)(mt * 16 + nl) * E_ + hcol + 2 * hf;
#pragma unroll
        for (int s = 0; s < 16; ++s) {
            qa_r[s] = *(const v2f*)(qr_p + 4 * s);
            qa_i[s] = *(const v2f*)(qi_p + 4 * s);
        }
    }

    v8f aor[4], aoi[4];   // output accumulators: 4 col-tiles of 16 over D=64
#pragma unroll
    for (int j = 0; j < 4; ++j) {
        aor[j] = (v8f){0.f, 0.f, 0.f, 0.f, 0.f, 0.f, 0.f, 0.f};
        aoi[j] = (v8f){0.f, 0.f, 0.f, 0.f, 0.f, 0.f, 0.f, 0.f};
    }
    float m_run[8], l_run[8];
#pragma unroll
    for (int g = 0; g < 8; ++g) { m_run[g] = -3.0e38f; l_run[g] = 0.f; }

    for (int kt = 0; kt < L_ / 16; ++kt) {
        // ---- scores tile: S = qr*kr^T + qi*ki^T (one shared f32 accumulator)
        v8f c = (v8f){0.f, 0.f, 0.f, 0.f, 0.f, 0.f, 0.f, 0.f};
        const float* kr_p = Kr + rowbase + (size_t)(kt * 16 + nl) * E_ + hcol + 2 * hf;
        const float* ki_p = Ki + rowbase + (size_t)(kt * 16 + nl) * E_ + hcol + 2 * hf;
#pragma unroll
        for (int s = 0; s < 16; ++s) {
            v2f kb_r = *(const v2f*)(kr_p + 4 * s);
            v2f kb_i = *(const v2f*)(ki_p + 4 * s);
            c = wmma_f32(qa_r[s], kb_r, c);
            c = wmma_f32(qa_i[s], kb_i, c);
        }

        // ---- online softmax (row reductions across 16-lane groups)
        float p[8];
#pragma unroll
        for (int g = 0; g < 8; ++g) {
            float t  = c[g] * scale;
            float rm = t;
            rm = fmaxf(rm, __shfl_xor(rm, 1, 32));
            rm = fmaxf(rm, __shfl_xor(rm, 2, 32));
            rm = fmaxf(rm, __shfl_xor(rm, 4, 32));
            rm = fmaxf(rm, __shfl_xor(rm, 8, 32));
            float mn = fmaxf(m_run[g], rm);
            float pe = __expf(t - mn);
            float rs = pe;
            rs += __shfl_xor(rs, 1, 32);
            rs += __shfl_xor(rs, 2, 32);
            rs += __shfl_xor(rs, 4, 32);
            rs += __shfl_xor(rs, 8, 32);
            float alpha = __expf(m_run[g] - mn);
            l_run[g] = l_run[g] * alpha + rs;
            m_run[g] = mn;
            p[g]     = pe;
#pragma unroll
            for (int j = 0; j < 4; ++j) { aor[j][g] *= alpha; aoi[j][g] *= alpha; }
        }

        // ---- re-layout P: C-layout regs -> LDS -> A-layout fragments
        __syncthreads();
#pragma unroll
        for (int g = 0; g < 8; ++g)
            lds[(g + 8 * hf) * 16 + nl] = p[g];
        __syncthreads();
        v2f pa[4];
#pragma unroll
        for (int s = 0; s < 4; ++s)
            pa[s] = *(const v2f*)&lds[nl * 16 + s * 4 + 2 * hf];

        // ---- O += P @ V  (M=16 queries, N=64 d-cols in 4 tiles, K=16 keys)
        const float* vr_p = Vr + rowbase + (size_t)(kt * 16) * E_ + hcol + nl;
        const float* vi_p = Vi + rowbase + (size_t)(kt * 16) * E_ + hcol + nl;
#pragma unroll
        for (int j = 0; j < 4; ++j) {
#pragma unroll
            for (int s = 0; s < 4; ++s) {
                const float* vr0 = vr_p + (size_t)(s * 4 + 2 * hf) * E_ + j * 16;
                const float* vi0 = vi_p + (size_t)(s * 4 + 2 * hf) * E_ + j * 16;
                v2f vbr; vbr.x = vr0[0]; vbr.y = vr0[E_];   // B(k,n)=V[key][d], k pair
                v2f vbi; vbi.x = vi0[0]; vbi.y = vi0[E_];
                aor[j] = wmma_f32(pa[s], vbr, aor[j]);
                aoi[j] = wmma_f32(pa[s], vbi, aoi[j]);
            }
        }
    }

    // ---- finalize: divide by softmax denominator, write heads-concat layout
#pragma unroll
    for (int g = 0; g < 8; ++g) {
        const float inv = 1.0f / l_run[g];
        const int   row = mt * 16 + g + 8 * hf;
        const size_t base = rowbase + (size_t)row * E_ + hcol;
#pragma unroll
        for (int j = 0; j < 4; ++j) {
            Or[base + j * 16 + nl] = aor[j][g] * inv;
            Oi[base + j * 16 + nl] = aoi[j][g] * inv;
        }
    }
}

// ---------------------------------------------------------------------------
// Host-side launcher
// Inputs (setup_inputs order):
//  0..5 : query_r, query_i, key_r, key_i, value_r, value_i     [B*L*E]
//  6..9 : Wq_r, Wq_i, bq_r, bq_i ; 10..13: Wk_* ; 14..17: Wv_* ; 18..21: Wo_*
// Output: [2, B, L, E] (real, imag)
// Workspace: 8 * B*L*E floats (~134 MB): Qr,Qi,Kr,Ki,Vr,Vi,AttnR,AttnI
// ---------------------------------------------------------------------------
extern "C" void kernel_launch(void* const* d_in, const int* in_sizes, int n_in,
                              void* d_out, int out_size, void* d_ws, size_t ws_size,
                              hipStream_t stream)
{
    (void)in_sizes; (void)n_in; (void)out_size; (void)ws_size;

    const float* q_r = (const float*)d_in[0];
    const float* q_i = (const float*)d_in[1];
    const float* k_r = (const float*)d_in[2];
    const float* k_i = (const float*)d_in[3];
    const float* v_r = (const float*)d_in[4];
    const float* v_i = (const float*)d_in[5];

    const float* Wq_r = (const float*)d_in[6];
    const float* Wq_i = (const float*)d_in[7];
    const float* bq_r = (const float*)d_in[8];
    const float* bq_i = (const float*)d_in[9];
    const float* Wk_r = (const float*)d_in[10];
    const float* Wk_i = (const float*)d_in[11];
    const float* bk_r = (const float*)d_in[12];
    const float* bk_i = (const float*)d_in[13];
    const float* Wv_r = (const float*)d_in[14];
    const float* Wv_i = (const float*)d_in[15];
    const float* bv_r = (const float*)d_in[16];
    const float* bv_i = (const float*)d_in[17];
    const float* Wo_r = (const float*)d_in[18];
    const float* Wo_i = (const float*)d_in[19];
    const float* bo_r = (const float*)d_in[20];
    const float* bo_i = (const float*)d_in[21];

    const size_t NBL = (size_t)B_ * L_ * E_;
    float* ws = (float*)d_ws;
    float* Qr = ws + 0 * NBL;
    float* Qi = ws + 1 * NBL;
    float* Kr = ws + 2 * NBL;
    float* Ki = ws + 3 * NBL;
    float* Vr = ws + 4 * NBL;
    float* Vi = ws + 5 * NBL;
    float* Ar = ws + 6 * NBL;
    float* Ai = ws + 7 * NBL;

    float* out_r = (float*)d_out;
    float* out_i = out_r + NBL;

    const dim3 blk(32);
    const dim3 gL(E_ / 64, (B_ * L_) / 16);     // 16 x 256 (16x64 tiles per wave)
    const dim3 gA(L_ / 16, B_ * H_);            // 128 x 32

    complex_linear_wmma<<<gL, blk, 0, stream>>>(q_r, q_i, Wq_r, Wq_i, bq_r, bq_i, Qr, Qi);
    complex_linear_wmma<<<gL, blk, 0, stream>>>(k_r, k_i, Wk_r, Wk_i, bk_r, bk_i, Kr, Ki);
    complex_linear_wmma<<<gL, blk, 0, stream>>>(v_r, v_i, Wv_r, Wv_i, bv_r, bv_i, Vr, Vi);
    cattn_flash_wmma  <<<gA, blk, 0, stream>>>(Qr, Qi, Kr, Ki, Vr, Vi, Ar, Ai);
    complex_linear_wmma<<<gL, blk, 0, stream>>>(Ar, Ai, Wo_r, Wo_i, bo_r, bo_i, out_r, out_i);
}